// VectorQuantizerEMA_56684978372674
// MI455X (gfx1250) — compile-verified
//
#include <hip/hip_runtime.h>
#include <hip/hip_bf16.h>

typedef __attribute__((ext_vector_type(2))) float v2f;
typedef __attribute__((ext_vector_type(8))) float v8f;

#define NUM_EMB 512
#define EMB_DIM 64
#define HW      4096        // 64*64
#define NROWS   131072      // 32*64*64

// ---------------------------------------------------------------------------
// Kernel 0: per-embedding bias  -0.5*||e_k||^2 ; zero histogram + loss accum
// ---------------------------------------------------------------------------
__global__ void vq_prep(const float* __restrict__ emb,
                        float* __restrict__ negHalf,
                        unsigned int* __restrict__ hist,
                        float* __restrict__ lossAcc) {
    int t = threadIdx.x;                       // blockDim = 512 = NUM_EMB
    const float* e = emb + (size_t)t * EMB_DIM;
    float s = 0.f;
#pragma unroll
    for (int i = 0; i < EMB_DIM; ++i) { float v = e[i]; s += v * v; }
    negHalf[t] = -0.5f * s;
    hist[t] = 0u;
    if (t == 0) *lossAcc = 0.f;
}

// ---------------------------------------------------------------------------
// Kernel 1: WMMA score GEMM + fused argmax + histogram + loss partials
//   grid = NROWS/128 blocks, 256 threads (8 waves); each wave owns 16 rows.
// ---------------------------------------------------------------------------
__global__ void __launch_bounds__(256) vq_main(const float* __restrict__ x,
                                               const float* __restrict__ emb,
                                               const float* __restrict__ negHalf,
                                               unsigned int* __restrict__ hist,
                                               float* __restrict__ lossAcc,
                                               float* __restrict__ outIdx) {
    const int lane    = threadIdx.x & 31;
    const int wave    = threadIdx.x >> 5;
    const int rowBase = (blockIdx.x * 8 + wave) * 16;   // 16 rows per wave
    const int b       = rowBase >> 12;                  // / 4096 (rows per image)
    const int hw      = rowBase & 4095;
    const int col     = lane & 15;                      // column within 16-group
    const int khalf   = (lane >> 4) * 2;                // k offset 0 or 2 (A/B layout)

    // ---- load A fragment (16x64 f32, 16 k-steps of 16x16x4) -------------
    // element (row, c) lives at x[b*C*HW + c*HW + (row % HW)] : rows contiguous
    const float* abase = x + (size_t)b * EMB_DIM * HW + hw + col;
    v2f afrag[16];
    float xnp = 0.f;                                    // partial ||x_row||^2
#pragma unroll
    for (int j = 0; j < 16; ++j) {
        float a0 = abase[(size_t)(4 * j + khalf)     * HW];
        float a1 = abase[(size_t)(4 * j + khalf + 1) * HW];
        afrag[j].x = a0; afrag[j].y = a1;
        xnp += a0 * a0 + a1 * a1;
    }
    // lane r and r+16 hold the two k-halves of row r -> combine
    float xnorm = xnp + __shfl_xor(xnp, 16, 32);        // lane r: ||x_{rowBase+r}||^2

    float bestScore[8];
    int   bestIdx[8];
#pragma unroll
    for (int r = 0; r < 8; ++r) { bestScore[r] = -__builtin_inff(); bestIdx[r] = 0; }

    // ---- sweep all 32 embedding tiles of 16 columns ---------------------
    for (int t = 0; t < NUM_EMB / 16; ++t) {
        const int embBase = t * 16;
        const float* bb = emb + (size_t)(embBase + col) * EMB_DIM + khalf;
        float bias = negHalf[embBase + col];            // per-column bias
        v8f acc;
#pragma unroll
        for (int r = 0; r < 8; ++r) acc[r] = bias;      // C := -0.5||e_n||^2
#pragma unroll
        for (int j = 0; j < 16; ++j) {
            v2f bfrag;
            bfrag.x = bb[4 * j];
            bfrag.y = bb[4 * j + 1];
            acc = __builtin_amdgcn_wmma_f32_16x16x4_f32(
                false, afrag[j], false, bfrag, (short)0, acc, false, false);
        }
        // running argmax (strict > keeps lowest index across tiles)
#pragma unroll
        for (int r = 0; r < 8; ++r) {
            if (acc[r] > bestScore[r]) { bestScore[r] = acc[r]; bestIdx[r] = embBase + col; }
        }
    }

    // ---- reduce each row slot across its 16-lane column group ----------
#pragma unroll
    for (int r = 0; r < 8; ++r) {
        float s = bestScore[r]; int i = bestIdx[r];
#pragma unroll
        for (int m = 8; m >= 1; m >>= 1) {
            float os = __shfl_xor(s, m, 32);
            int   oi = __shfl_xor(i, m, 32);
            if (os > s || (os == s && oi < i)) { s = os; i = oi; }
        }
        bestScore[r] = s; bestIdx[r] = i;               // uniform in each half
    }

    // lanes 0..7 take rows 0..7 (lower half), lanes 16..23 take rows 8..15
    float myScore = bestScore[0]; int myIdx = bestIdx[0];
#pragma unroll
    for (int r = 1; r < 8; ++r) {
        if ((lane & 15) == r) { myScore = bestScore[r]; myIdx = bestIdx[r]; }
    }
    const int  myRow  = (lane < 16) ? (lane & 7) : 8 + (lane & 7);
    const bool active = ((lane & 15) < 8);
    float myXnorm = __shfl(xnorm, myRow, 32);           // norm held in lane myRow

    float dist2 = myXnorm - 2.0f * myScore;             // ||x - e*||^2 (algebraic)
    if (active) {
        outIdx[rowBase + myRow] = (float)myIdx;
        atomicAdd(&hist[myIdx], 1u);
    }
    // one loss atomic per wave
    float contrib = active ? dist2 : 0.f;
#pragma unroll
    for (int m = 16; m >= 1; m >>= 1) contrib += __shfl_xor(contrib, m, 32);
    if (lane == 0) atomicAdd(lossAcc, contrib);
}

// ---------------------------------------------------------------------------
// Kernel 2: finalize loss + perplexity (1 block, 512 threads)
// ---------------------------------------------------------------------------
__global__ void vq_final(const unsigned int* __restrict__ hist,
                         const float* __restrict__ lossAcc,
                         float* __restrict__ out) {
    __shared__ float sm[NUM_EMB];
    int t = threadIdx.x;
    float p = (float)hist[t] / (float)NROWS;
    sm[t] = p * __logf(p + 1e-10f);
    __syncthreads();
    for (int s = NUM_EMB / 2; s > 0; s >>= 1) {
        if (t < s) sm[t] += sm[t + s];
        __syncthreads();
    }
    if (t == 0) {
        out[1] = __expf(-sm[0]);                                  // perplexity
        out[0] = 0.25f * (*lossAcc) / (float)((size_t)NROWS * EMB_DIM); // loss
    }
}

// ---------------------------------------------------------------------------
extern "C" void kernel_launch(void* const* d_in, const int* in_sizes, int n_in,
                              void* d_out, int out_size, void* d_ws, size_t ws_size,
                              hipStream_t stream) {
    const float* x   = (const float*)d_in[0];   // [32,64,64,64] f32
    const float* emb = (const float*)d_in[1];   // [512,64] f32
    float* out = (float*)d_out;                 // [loss, perplexity, indices...]

    float*        negHalf = (float*)d_ws;                              // 512 f32
    unsigned int* hist    = (unsigned int*)((char*)d_ws + 512 * 4);    // 512 u32
    float*        lossAcc = (float*)((char*)d_ws + 1024 * 4);          // 1 f32

    vq_prep <<<1, NUM_EMB, 0, stream>>>(emb, negHalf, hist, lossAcc);
    vq_main <<<NROWS / 128, 256, 0, stream>>>(x, emb, negHalf, hist, lossAcc, out + 2);
    vq_final<<<1, NUM_EMB, 0, stream>>>(hist, lossAcc, out);
}